// fk_layer_44710609551762
// MI455X (gfx1250) — compile-verified
//
#include <hip/hip_runtime.h>
#include <stdint.h>

#define NJ 19
#define ROWF 76          // floats per joint row (19 quats * 4)
#define ROWB 304         // bytes per row
#define TPB 256          // 8 waves of 32
#define ROOT_SCALE 200.0f

typedef float v4f __attribute__((ext_vector_type(4)));   // native vector for NT store

__global__ __launch_bounds__(TPB) void fk_kernel(
    const float* __restrict__ root,
    const float* __restrict__ joint,
    const float* __restrict__ offsets,
    float* __restrict__ out)
{
    __shared__ float smem[TPB * ROWF];   // 77824 B -> 4 workgroups/WGP fit in 320KB LDS

    const int T    = threadIdx.x;
    const int lane = T & 31;
    const int wave = T >> 5;
    const long long blockRow = (long long)blockIdx.x * TPB;
    const long long b = blockRow + T;

    // ---- CDNA5 async stage: each wave DMAs its own 32 rows (9728 B) of
    //      quaternion data global -> LDS. 32 lanes x b128 = coalesced 512 B
    //      per issue, tracked on ASYNCcnt; no cross-wave barrier needed.
    //      Streaming input -> non-temporal load hint. ----
    {
        const char*    gbase    = (const char*)joint + blockRow * (long long)ROWB;
        const uint32_t ldsBase  = (uint32_t)(uintptr_t)(&smem[0]);
        const uint32_t waveByte = (uint32_t)(wave * 32 * ROWB);
#pragma unroll
        for (int i = 0; i < 19; ++i) {
            const uint32_t goff = waveByte + (uint32_t)((i * 32 + lane) * 16);
            const uint32_t dst  = ldsBase + goff;
            asm volatile("global_load_async_to_lds_b128 %0, %1, %2 th:TH_LOAD_NT"
                         :: "v"(dst), "v"(goff), "s"(gbase)
                         : "memory");
        }
        asm volatile("s_wait_asynccnt 0x0" ::: "memory");
    }

    // Uniform per-joint offset translations -> scalar (SGPR) loads.
    float tr0[NJ], tr1[NJ], tr2[NJ];
#pragma unroll
    for (int j = 0; j < NJ; ++j) {
        tr0[j] = offsets[j * 16 + 3];
        tr1[j] = offsets[j * 16 + 7];
        tr2[j] = offsets[j * 16 + 11];
    }

    const float rs0 = root[b * 3 + 0] * ROOT_SCALE;
    const float rs1 = root[b * 3 + 1] * ROOT_SCALE;
    const float rs2 = root[b * 3 + 2] * ROOT_SCALE;

    constexpr int PAR[NJ] = {-1,0,1,2,3,2,5,6,7,2,9,10,11,0,13,14,0,16,17};

    float Mr[NJ][9];   // fully unrolled constant indices -> registers (SROA)
    float Mt[NJ][3];
    const float* q = &smem[T * ROWF];
    float* orow    = out + b * (long long)ROWF;

#pragma unroll
    for (int j = 0; j < NJ; ++j) {
        // quat -> rotation (reference formulation, un-normalized quats)
        const float qw = q[4*j+0], qx = q[4*j+1], qy = q[4*j+2], qz = q[4*j+3];
        const float s2 = 2.0f / (qw*qw + qx*qx + qy*qy + qz*qz);
        const float xx = s2*qx*qx, yy = s2*qy*qy, zz = s2*qz*qz;
        const float xy = s2*qx*qy, xz = s2*qx*qz, yz = s2*qy*qz;
        const float wx = s2*qw*qx, wy = s2*qw*qy, wz = s2*qw*qz;
        const float r00 = 1.0f-(yy+zz), r01 = xy-wz,        r02 = xz+wy;
        const float r10 = xy+wz,        r11 = 1.0f-(xx+zz), r12 = yz-wx;
        const float r20 = xz-wy,        r21 = yz+wx,        r22 = 1.0f-(xx+yy);

        // RT_j = [R | R*t]  (offsets are identity rotation + translation)
        const float tx = tr0[j], ty = tr1[j], tz = tr2[j];
        const float ttx = r00*tx + r01*ty + r02*tz;
        const float tty = r10*tx + r11*ty + r12*tz;
        const float ttz = r20*tx + r21*ty + r22*tz;

        if (j == 0) {
            Mr[0][0]=r00; Mr[0][1]=r01; Mr[0][2]=r02;
            Mr[0][3]=r10; Mr[0][4]=r11; Mr[0][5]=r12;
            Mr[0][6]=r20; Mr[0][7]=r21; Mr[0][8]=r22;
            Mt[0][0]=ttx; Mt[0][1]=tty; Mt[0][2]=ttz;
        } else {
            const int p = PAR[j];
            // M_j = M_p * RT_j (affine 3x4)
            Mr[j][0]=Mr[p][0]*r00 + Mr[p][1]*r10 + Mr[p][2]*r20;
            Mr[j][1]=Mr[p][0]*r01 + Mr[p][1]*r11 + Mr[p][2]*r21;
            Mr[j][2]=Mr[p][0]*r02 + Mr[p][1]*r12 + Mr[p][2]*r22;
            Mr[j][3]=Mr[p][3]*r00 + Mr[p][4]*r10 + Mr[p][5]*r20;
            Mr[j][4]=Mr[p][3]*r01 + Mr[p][4]*r11 + Mr[p][5]*r21;
            Mr[j][5]=Mr[p][3]*r02 + Mr[p][4]*r12 + Mr[p][5]*r22;
            Mr[j][6]=Mr[p][6]*r00 + Mr[p][7]*r10 + Mr[p][8]*r20;
            Mr[j][7]=Mr[p][6]*r01 + Mr[p][7]*r11 + Mr[p][8]*r21;
            Mr[j][8]=Mr[p][6]*r02 + Mr[p][7]*r12 + Mr[p][8]*r22;
            Mt[j][0]=Mr[p][0]*ttx + Mr[p][1]*tty + Mr[p][2]*ttz + Mt[p][0];
            Mt[j][1]=Mr[p][3]*ttx + Mr[p][4]*tty + Mr[p][5]*ttz + Mt[p][1];
            Mt[j][2]=Mr[p][6]*ttx + Mr[p][7]*tty + Mr[p][8]*ttz + Mt[p][2];
        }

        v4f o;
        o.x = Mt[j][0] + rs0;
        o.y = Mt[j][1] + rs1;
        o.z = Mt[j][2] + rs2;
        o.w = 1.0f;
        // Streaming output: 16B-aligned non-temporal b128 store.
        __builtin_nontemporal_store(o, reinterpret_cast<v4f*>(orow + j * 4));
    }
}

extern "C" void kernel_launch(void* const* d_in, const int* in_sizes, int n_in,
                              void* d_out, int out_size, void* d_ws, size_t ws_size,
                              hipStream_t stream) {
    const float* root    = (const float*)d_in[0];   // (B, 3)
    const float* joint   = (const float*)d_in[1];   // (B, 76)
    const float* offsets = (const float*)d_in[2];   // (19, 4, 4)
    float* out = (float*)d_out;                     // (B, 19, 4)

    const int B = in_sizes[0] / 3;                  // 262144
    dim3 grid((B + TPB - 1) / TPB);
    dim3 block(TPB);
    hipLaunchKernelGGL(fk_kernel, grid, block, 0, stream, root, joint, offsets, out);
}